// GATLayer_60292750901519
// MI455X (gfx1250) — compile-verified
//
#include <hip/hip_runtime.h>
#include <hip/hip_bf16.h>

// GAT layer on gfx1250 (MI455X, wave32).
//
// Phases:
//  K1: h0 = x @ fc_w^T via V_WMMA_F32_16X16X4_F32 (exact f32 matrix path).
//  K2: s_src[i] = h0[i,:]·a_w[:64], s_dst[i] = h0[i,:]·a_w[64:], h_sum[i]=0.
//  K3: per-edge h = exp(leakyrelu(s_src[s]+s_dst[d]+b)); atomic add into h_sum[s].
//  K4: alpha = h/h_sum[s]; out[s,:] += adj[e]*alpha*h0[d,:]  (64 threads/edge).

typedef __attribute__((ext_vector_type(2))) float v2f;
typedef __attribute__((ext_vector_type(8))) float v8f;

#define NFEAT 128
#define NHID  64
#define WP_STRIDE 72  // v2f units; 144 dwords -> +2 kp rows shifts banks by 32

// ---------------------------------------------------------------------------
// K1: GEMM h0[M,64] = x[M,128] @ fc_w[64,128]^T using WMMA f32 16x16x4.
// One wave per 16-row strip, 8 waves (256 thr) per block.
// LDS holds fc_w as K-pair-interleaved float2: wP[k/2][n] = {w[n][k], w[n][k+1]},
// so each B fragment is ONE aligned ds_load_b64 straight into an even VGPR
// pair (WMMA operand requirement) — no repacking movs.
// A layout (ISA 7.12.2, 32-bit A 16x4): lane<16 -> M=lane, K={k,k+1};
// lane>=16 -> M=lane-16, K={k+2,k+3}. B mirrors the half-split at K+2.
// Row stride 72 v2f pads banks; lanes 16-31 read row kp+1, 144 dwords past lanes 0-15.
// Also zeroes the `out` region of d_out (same address pattern as h0 stores).
// ---------------------------------------------------------------------------
__global__ __launch_bounds__(256) void gat_gemm_h0(
    const float* __restrict__ x, const float* __restrict__ fc_w,
    float* __restrict__ h0, float* __restrict__ out_zero, int n_nodes) {
  __shared__ v2f wP[(NFEAT / 2) * WP_STRIDE];  // 36 KB
  const int tid = threadIdx.x;
  for (int i = tid; i < NHID * NFEAT; i += 256) {
    int n = i >> 7;   // fc_w row (hidden unit)
    int k = i & 127;  // fc_w col (feature)
    ((float*)&wP[(k >> 1) * WP_STRIDE + n])[k & 1] = fc_w[i];
  }
  __syncthreads();

  const int wave = tid >> 5;
  const int lane = tid & 31;
  const int row0 = (blockIdx.x * 8 + wave) * 16;
  if (row0 >= n_nodes) return;  // wave-uniform: EXEC stays all-1s for WMMA

  const int half = lane >> 4;   // 0: lanes 0-15, 1: lanes 16-31
  const int l15  = lane & 15;
  const float* xrow = x + (size_t)(row0 + l15) * NFEAT + half * 2;

  v8f c0 = {}, c1 = {}, c2 = {}, c3 = {};
  for (int k = 0; k < NFEAT; k += 4) {
    v2f a;
    a.x = xrow[k];
    a.y = xrow[k + 1];
    const v2f* wrow = &wP[(k / 2 + half) * WP_STRIDE + l15];
    v2f b0 = wrow[0];
    v2f b1 = wrow[16];
    v2f b2 = wrow[32];
    v2f b3 = wrow[48];
    c0 = __builtin_amdgcn_wmma_f32_16x16x4_f32(false, a, false, b0, (short)0, c0, false, false);
    c1 = __builtin_amdgcn_wmma_f32_16x16x4_f32(false, a, false, b1, (short)0, c1, false, false);
    c2 = __builtin_amdgcn_wmma_f32_16x16x4_f32(false, a, false, b2, (short)0, c2, false, false);
    c3 = __builtin_amdgcn_wmma_f32_16x16x4_f32(false, a, false, b3, (short)0, c3, false, false);
  }

  // C/D layout: VGPR r -> M = row0 + r + 8*half, N = l15 (+16*tile).
#pragma unroll
  for (int r = 0; r < 8; ++r) {
    int m = row0 + r + half * 8;
    if (m < n_nodes) {
      size_t base = (size_t)m * NHID + l15;
      h0[base +  0] = c0[r];
      h0[base + 16] = c1[r];
      h0[base + 32] = c2[r];
      h0[base + 48] = c3[r];
      out_zero[base +  0] = 0.0f;
      out_zero[base + 16] = 0.0f;
      out_zero[base + 32] = 0.0f;
      out_zero[base + 48] = 0.0f;
    }
  }
}

// ---------------------------------------------------------------------------
// K2: per-node attention scores + zero h_sum.
// ---------------------------------------------------------------------------
__global__ __launch_bounds__(256) void gat_node_scores(
    const float* __restrict__ h0, const float* __restrict__ a_w,
    float* __restrict__ s_src, float* __restrict__ s_dst,
    float* __restrict__ h_sum, int n_nodes) {
  int i = blockIdx.x * blockDim.x + threadIdx.x;
  if (i >= n_nodes) return;
  const float* row = h0 + (size_t)i * NHID;
  float ss = 0.0f, sd = 0.0f;
#pragma unroll
  for (int j = 0; j < NHID; ++j) {
    float v = row[j];
    ss += v * a_w[j];
    sd += v * a_w[NHID + j];
  }
  s_src[i] = ss;
  s_dst[i] = sd;
  h_sum[i] = 0.0f;
}

// ---------------------------------------------------------------------------
// K3: per-edge exp(leakyrelu) + segment-sum over src via hw float atomics.
// ---------------------------------------------------------------------------
__global__ __launch_bounds__(256) void gat_edge_exp(
    const long long* __restrict__ ei, const float* __restrict__ s_src,
    const float* __restrict__ s_dst, const float* __restrict__ a_b,
    float* __restrict__ h_buf, float* __restrict__ h_sum, int n_edges) {
  int e = blockIdx.x * blockDim.x + threadIdx.x;
  if (e >= n_edges) return;
  int s = (int)ei[e];
  int d = (int)ei[(size_t)n_edges + e];
  float v = s_src[s] + s_dst[d] + a_b[0];
  v = (v >= 0.0f) ? v : 0.05f * v;   // LeakyReLU(0.05)
  float h = expf(v);
  h_buf[e] = h;
  unsafeAtomicAdd(&h_sum[s], h);
}

// ---------------------------------------------------------------------------
// K4: alpha = h/h_sum[src]; out[src,:] += adj*alpha*h0[dst,:].
// 64 threads per edge (coalesced h0[dst] gather + consecutive f32 atomics).
// ---------------------------------------------------------------------------
__global__ __launch_bounds__(256) void gat_edge_scatter(
    const long long* __restrict__ ei, const float* __restrict__ adj,
    const float* __restrict__ h0, const float* __restrict__ h_sum,
    const float* __restrict__ h_buf, float* __restrict__ alpha_out,
    float* __restrict__ out, int n_edges) {
  long long t = (long long)blockIdx.x * blockDim.x + threadIdx.x;
  int e = (int)(t >> 6);
  int j = (int)(t & 63);
  if (e >= n_edges) return;
  int s = (int)ei[e];
  int d = (int)ei[(size_t)n_edges + e];
  float alpha = h_buf[e] / h_sum[s];
  if (j == 0) alpha_out[e] = alpha;
  float val = adj[e] * alpha;
  float contrib = val * h0[(size_t)d * NHID + j];
  unsafeAtomicAdd(&out[(size_t)s * NHID + j], contrib);
}

extern "C" void kernel_launch(void* const* d_in, const int* in_sizes, int n_in,
                              void* d_out, int out_size, void* d_ws, size_t ws_size,
                              hipStream_t stream) {
  const float*     x    = (const float*)d_in[0];
  const long long* ei   = (const long long*)d_in[1];  // int64 [2, E]
  const float*     adj  = (const float*)d_in[2];
  const float*     fc_w = (const float*)d_in[3];
  const float*     a_w  = (const float*)d_in[4];
  const float*     a_b  = (const float*)d_in[5];

  const int n_nodes = in_sizes[0] / NFEAT;   // 100000
  const int n_edges = in_sizes[2];           // 1600000

  // Output: [ out (n_nodes*64) | alpha (n_edges) ]
  float* out_ptr   = (float*)d_out;
  float* alpha_ptr = out_ptr + (size_t)n_nodes * NHID;

  // Workspace: h0 | s_src | s_dst | h_sum | h_buf  (~33 MB)
  float* h0    = (float*)d_ws;
  float* s_src = h0 + (size_t)n_nodes * NHID;
  float* s_dst = s_src + n_nodes;
  float* h_sum = s_dst + n_nodes;
  float* h_buf = h_sum + n_nodes;

  // K1: GEMM. 16-row strips, 8 strips per block.
  int strips = (n_nodes + 15) / 16;
  int blocks1 = (strips + 7) / 8;
  gat_gemm_h0<<<blocks1, 256, 0, stream>>>(x, fc_w, h0, out_ptr, n_nodes);

  // K2: node scores + zero h_sum.
  int blocks2 = (n_nodes + 255) / 256;
  gat_node_scores<<<blocks2, 256, 0, stream>>>(h0, a_w, s_src, s_dst, h_sum, n_nodes);

  // K3: edge exp + segment sum.
  int blocks3 = (n_edges + 255) / 256;
  gat_edge_exp<<<blocks3, 256, 0, stream>>>(ei, s_src, s_dst, a_b, h_buf, h_sum, n_edges);

  // K4: normalize + weighted scatter (64 threads per edge).
  long long tot = (long long)n_edges * NHID;
  int blocks4 = (int)((tot + 255) / 256);
  gat_edge_scatter<<<blocks4, 256, 0, stream>>>(ei, adj, h0, h_sum, h_buf,
                                                alpha_ptr, out_ptr, n_edges);
}